// MultiHeadAttention_9277129360125
// MI455X (gfx1250) — compile-verified
//
#include <hip/hip_runtime.h>
#include <math.h>

typedef float v2f __attribute__((ext_vector_type(2)));
typedef float v8f __attribute__((ext_vector_type(8)));

#define B_   32
#define E_   2048
#define H_   16
#define D_   128
#define KV_  2048

// ---------------------------------------------------------------------------
// GEMM C(MxN) = A(MxK) @ W(KxN) + bias, fp32, via V_WMMA_F32_16X16X4_F32.
// One wave (32 threads) per 16x16 output tile. grid = (N/16, M/16).
//
// A fragment (16x4, 2 VGPRs): lanes 0-15 M=lane (K=k,k+1), lanes 16-31 M=lane-16 (K=k+2,k+3)
// B fragment (4x16, 2 VGPRs): lanes 0-15 col=lane row K=k/k+1, lanes 16-31 row K=k+2/k+3
// C/D (16x16, 8 VGPRs): vgpr r, lanes 0-15 -> M=r, lanes 16-31 -> M=8+r; N=lane&15
// ---------------------------------------------------------------------------
__global__ __launch_bounds__(32)
void gemm_wmma_f32(const float* __restrict__ A, const float* __restrict__ W,
                   const float* __restrict__ bias, float* __restrict__ C,
                   int K, int N) {
    const int nt   = blockIdx.x;
    const int mt   = blockIdx.y;
    const int lane = threadIdx.x & 31;
    const int half = lane >> 4;       // 0: K pair {k,k+1}, 1: {k+2,k+3}
    const int l16  = lane & 15;
    const int m0 = mt * 16, n0 = nt * 16;

    const float* Arow = A + (size_t)(m0 + l16) * K;   // this lane's A row
    const float* Wcol = W + n0 + l16;                 // this lane's W column

    v8f c = {};
#pragma unroll 4
    for (int k = 0; k < K; k += 4) {
        const int ka = k + 2 * half;
        v2f a, b;
        a.x = Arow[ka];
        a.y = Arow[ka + 1];
        b.x = Wcol[(size_t)ka * N];
        b.y = Wcol[(size_t)(ka + 1) * N];
        c = __builtin_amdgcn_wmma_f32_16x16x4_f32(
                /*neg_a=*/false, a, /*neg_b=*/false, b,
                /*c_mod=*/(short)0, c, /*reuse_a=*/false, /*reuse_b=*/false);
    }

    const float bv = bias[n0 + l16];
#pragma unroll
    for (int r = 0; r < 8; ++r) {
        const int m = (half ? 8 : 0) + r;
        C[(size_t)(m0 + m) * N + n0 + l16] = c[r] + bv;
    }
}

// ---------------------------------------------------------------------------
// Fused attention + cache concat. One block (8 waves / 256 thr) per (b,h).
// Pass 1 streams K rows: float4 per lane (B128), fused store to Kc output,
// wave32 shuffle reduction for the score. Softmax in LDS. Pass 2 streams V
// rows: fused store to Vc + context accumulation, LDS cross-wave reduce.
// ---------------------------------------------------------------------------
__global__ __launch_bounds__(256)
void attn_fused(const float* __restrict__ qkv,     // (B, 3E): [Q | K | V]
                const float* __restrict__ kcache,  // (B,H,KV,D)
                const float* __restrict__ vcache,  // (B,H,KV,D)
                const float* __restrict__ mask,    // (B,KV)
                float* __restrict__ Kc,            // (B,H,KV+1,D)
                float* __restrict__ Vc,            // (B,H,KV+1,D)
                float* __restrict__ ctx)           // (B,E)
{
    const int bh   = blockIdx.x;       // 0..B*H-1
    const int b    = bh >> 4;
    const int h    = bh & (H_ - 1);
    const int t    = threadIdx.x;
    const int lane = t & 31;
    const int w    = t >> 5;           // wave id 0..7

    __shared__ float s_scores[KV_ + 1];
    __shared__ float s_red[256];
    __shared__ float s_ctx[8][D_];

    const float scale = 0.088388347648318447f;  // 1/sqrt(128)

    const float* qrow = qkv + (size_t)b * 3 * E_ + h * D_;
    const float* knew = qrow + E_;
    const float* vnew = qrow + 2 * E_;
    const float4 q4 = *(const float4*)(qrow + lane * 4);

    const float* kbase  = kcache + (size_t)bh * KV_ * D_;
    const float* vbase  = vcache + (size_t)bh * KV_ * D_;
    float*       kcb    = Kc + (size_t)bh * (KV_ + 1) * D_;
    float*       vcb    = Vc + (size_t)bh * (KV_ + 1) * D_;
    const float* mrow   = mask + (size_t)b * KV_;

    // ---- pass 1: scores + fused Kc write -------------------------------
    for (int k = w; k < KV_ + 1; k += 8) {
        float4 k4 = (k < KV_) ? *(const float4*)(kbase + (size_t)k * D_ + lane * 4)
                              : *(const float4*)(knew + lane * 4);
        *(float4*)(kcb + (size_t)k * D_ + lane * 4) = k4;   // concat copy, same pass
        float p = q4.x * k4.x + q4.y * k4.y + q4.z * k4.z + q4.w * k4.w;
#pragma unroll
        for (int off = 16; off > 0; off >>= 1) p += __shfl_xor(p, off, 32);
        if (lane == 0) {
            const float mv = (k < KV_) ? mrow[k] : 1.0f;
            s_scores[k] = (mv == 0.0f) ? -INFINITY : p * scale;
        }
    }
    __syncthreads();

    // ---- softmax: max ---------------------------------------------------
    float lmax = -INFINITY;
    for (int k = t; k < KV_ + 1; k += 256) lmax = fmaxf(lmax, s_scores[k]);
    s_red[t] = lmax;
    __syncthreads();
#pragma unroll
    for (int s = 128; s > 0; s >>= 1) {
        if (t < s) s_red[t] = fmaxf(s_red[t], s_red[t + s]);
        __syncthreads();
    }
    const float gmax = s_red[0];
    __syncthreads();

    // ---- softmax: exp + sum --------------------------------------------
    float lsum = 0.0f;
    for (int k = t; k < KV_ + 1; k += 256) {
        const float e = __expf(s_scores[k] - gmax);
        s_scores[k] = e;
        lsum += e;
    }
    s_red[t] = lsum;
    __syncthreads();
#pragma unroll
    for (int s = 128; s > 0; s >>= 1) {
        if (t < s) s_red[t] += s_red[t + s];
        __syncthreads();
    }
    const float inv_sum = 1.0f / s_red[0];
    __syncthreads();

    // ---- pass 2: context + fused Vc write ------------------------------
    float4 acc = {0.f, 0.f, 0.f, 0.f};
    for (int k = w; k < KV_ + 1; k += 8) {
        float4 v4 = (k < KV_) ? *(const float4*)(vbase + (size_t)k * D_ + lane * 4)
                              : *(const float4*)(vnew + lane * 4);
        *(float4*)(vcb + (size_t)k * D_ + lane * 4) = v4;   // concat copy, same pass
        const float p = s_scores[k];
        acc.x += p * v4.x; acc.y += p * v4.y; acc.z += p * v4.z; acc.w += p * v4.w;
    }
    *(float4*)(&s_ctx[w][lane * 4]) = acc;
    __syncthreads();

    if (t < 32) {
        float4 sum = {0.f, 0.f, 0.f, 0.f};
#pragma unroll
        for (int ww = 0; ww < 8; ++ww) {
            const float4 v = *(const float4*)(&s_ctx[ww][t * 4]);
            sum.x += v.x; sum.y += v.y; sum.z += v.z; sum.w += v.w;
        }
        sum.x *= inv_sum; sum.y *= inv_sum; sum.z *= inv_sum; sum.w *= inv_sum;
        *(float4*)(ctx + (size_t)b * E_ + h * D_ + t * 4) = sum;
    }
}

// ---------------------------------------------------------------------------
// mask_new = concat(mask, ones(B,1))
// ---------------------------------------------------------------------------
__global__ void mask_concat(const float* __restrict__ mask, float* __restrict__ out) {
    const int i = blockIdx.x * blockDim.x + threadIdx.x;
    const int total = B_ * (KV_ + 1);
    if (i >= total) return;
    const int b = i / (KV_ + 1);
    const int k = i - b * (KV_ + 1);
    out[i] = (k < KV_) ? mask[(size_t)b * KV_ + k] : 1.0f;
}

// ---------------------------------------------------------------------------
extern "C" void kernel_launch(void* const* d_in, const int* in_sizes, int n_in,
                              void* d_out, int out_size, void* d_ws, size_t ws_size,
                              hipStream_t stream) {
    const float* X      = (const float*)d_in[0];  // (B,1,E)
    const float* kcache = (const float*)d_in[1];  // (B,H,KV,D)
    const float* vcache = (const float*)d_in[2];  // (B,H,KV,D)
    const float* mask   = (const float*)d_in[3];  // (B,KV)
    const float* Wqkv   = (const float*)d_in[4];  // (E,3E)
    const float* bqkv   = (const float*)d_in[5];  // (3E,)
    const float* Wo     = (const float*)d_in[6];  // (E,E)
    const float* bo     = (const float*)d_in[7];  // (E,)

    // d_out layout (return order, flat): out | Kc | Vc | mask_new
    float* out_y    = (float*)d_out;                                 // B*E
    float* out_Kc   = out_y + (size_t)B_ * E_;                       // B*H*(KV+1)*D
    float* out_Vc   = out_Kc + (size_t)B_ * H_ * (KV_ + 1) * D_;
    float* out_mask = out_Vc + (size_t)B_ * H_ * (KV_ + 1) * D_;     // B*(KV+1)

    // workspace: qkv (B x 3E) then ctx (B x E)
    float* ws_qkv = (float*)d_ws;
    float* ws_ctx = ws_qkv + (size_t)B_ * 3 * E_;

    // 1) QKV projection: (32 x 2048) @ (2048 x 6144) + bqkv
    {
        dim3 grid((3 * E_) / 16, B_ / 16);   // (384, 2)
        gemm_wmma_f32<<<grid, 32, 0, stream>>>(X, Wqkv, bqkv, ws_qkv, E_, 3 * E_);
    }

    // 2) fused attention + KV-cache concat
    attn_fused<<<B_ * H_, 256, 0, stream>>>(ws_qkv, kcache, vcache, mask,
                                            out_Kc, out_Vc, ws_ctx);

    // 3) output projection: (32 x 2048) @ (2048 x 2048) + bo
    {
        dim3 grid(E_ / 16, B_ / 16);         // (128, 2)
        gemm_wmma_f32<<<grid, 32, 0, stream>>>(ws_ctx, Wo, bo, out_y, E_, E_);
    }

    // 4) mask concat
    {
        const int total = B_ * (KV_ + 1);
        mask_concat<<<(total + 255) / 256, 256, 0, stream>>>(mask, out_mask);
    }
}